// SplitModule_54254026883542
// MI455X (gfx1250) — compile-verified
//
#include <hip/hip_runtime.h>

typedef float v2f __attribute__((ext_vector_type(2)));
typedef float v8f __attribute__((ext_vector_type(8)));

#define DFEAT 64

// ---------------- Phase 1: build expert-sorted permutation ----------------

__global__ void zero_meta_kernel(int* ws) {
  // counts[0..15], offsets[16..31], cursors[32..47], spare[48..63]
  ws[threadIdx.x] = 0;
}

__global__ void count_kernel(const int* __restrict__ inds, int* __restrict__ counts, int n) {
  int i = blockIdx.x * blockDim.x + threadIdx.x;
  if (i < n) atomicAdd(&counts[inds[i]], 1);
}

__global__ void scan_kernel(const int* __restrict__ counts, int* __restrict__ offsets,
                            int* __restrict__ cursors, int E) {
  if (threadIdx.x == 0) {
    int acc = 0;
    for (int e = 0; e < E; ++e) {
      offsets[e] = acc;
      cursors[e] = acc;
      acc += counts[e];
    }
    offsets[E] = acc;
  }
}

__global__ void scatter_kernel(const int* __restrict__ inds, int* __restrict__ cursors,
                               int* __restrict__ perm, int n) {
  int i = blockIdx.x * blockDim.x + threadIdx.x;
  if (i < n) {
    int pos = atomicAdd(&cursors[inds[i]], 1);
    perm[pos] = i;
  }
}

// ---------------- Phase 2: grouped GEMM via fp32 WMMA ----------------
// One wave32 owns a tile of 16 expert-sorted tokens x 64 output features.
// A: 16x64 fp32 (token features), B: 64x64 fp32 (W[e]), C: 4x (16x16) f32 accum.
// 64 x v_wmma_f32_16x16x4_f32 per tile iteration.

__global__ __launch_bounds__(128) void moe_wmma_kernel(
    const float* __restrict__ X, const int* __restrict__ inds,
    const float* __restrict__ W, const float* __restrict__ Bv,
    const int* __restrict__ perm, float* __restrict__ out, int n)
{
  const int lane = threadIdx.x & 31;
  const int wave = threadIdx.x >> 5;
  const int tile = blockIdx.x * 4 + wave;
  const int base = tile * 16;
  if (base >= n) return;                 // wave-uniform branch: EXEC stays full

  const int m    = lane & 15;            // matrix row index handled by this lane
  const int half = lane >> 4;            // 0: K slot {0,1}, 1: K slot {2,3}

  int  pos    = base + m;
  bool pvalid = pos < n;
  int  posc   = pvalid ? pos : (n - 1);  // clamp (select, no divergence)
  int  tok    = perm[posc];
  int  ind    = pvalid ? inds[tok] : -1;

  // Token index of each C/D row this lane stores (row = r + 8*half).
  int tokrow[8];
#pragma unroll
  for (int r = 0; r < 8; ++r)
    tokrow[r] = __builtin_amdgcn_ds_bpermute(((r + 8 * half) << 2), tok);

  // A fragments (ISA 16x4 fp32 layout): a[k] = X[tok(m)][4k + 2*half + {0,1}]
  v2f a[16];
  const float* xrow = X + (size_t)tok * DFEAT + half * 2;
#pragma unroll
  for (int k = 0; k < 16; ++k)
    a[k] = *(const v2f*)(xrow + 4 * k);

  unsigned validmask = __builtin_amdgcn_ballot_w32(ind >= 0) & 0xFFFFu;
  unsigned processed = 0;
  const v8f vzero = {0.f, 0.f, 0.f, 0.f, 0.f, 0.f, 0.f, 0.f};

  // Uniform loop over distinct experts in this tile (1 iter after sorting,
  // 2 only on the <=E-1 expert-boundary tiles).
  while (true) {
    unsigned todo = validmask & ~processed;
    if (todo == 0) break;
    int      firstrow = __builtin_ctz(todo);
    int      e        = __builtin_amdgcn_readlane(ind, firstrow);  // uniform
    unsigned rows     = __builtin_amdgcn_ballot_w32(ind == e) & 0xFFFFu;
    processed |= rows;

    const float* wbase = W + (size_t)e * DFEAT * DFEAT;

    v8f acc0 = vzero, acc1 = vzero, acc2 = vzero, acc3 = vzero;
#pragma unroll
    for (int k = 0; k < 16; ++k) {
      // B fragment: lane holds W[e][4k + 2*half + v][nt*16 + m], v in {0,1}
      const float* wp = wbase + (size_t)(4 * k + 2 * half) * DFEAT + m;
      v2f b0, b1, b2, b3;
      b0.x = wp[0];   b0.y = wp[DFEAT];
      b1.x = wp[16];  b1.y = wp[DFEAT + 16];
      b2.x = wp[32];  b2.y = wp[DFEAT + 32];
      b3.x = wp[48];  b3.y = wp[DFEAT + 48];
      acc0 = __builtin_amdgcn_wmma_f32_16x16x4_f32(false, a[k], false, b0, (short)0, acc0, false, false);
      acc1 = __builtin_amdgcn_wmma_f32_16x16x4_f32(false, a[k], false, b1, (short)0, acc1, false, false);
      acc2 = __builtin_amdgcn_wmma_f32_16x16x4_f32(false, a[k], false, b2, (short)0, acc2, false, false);
      acc3 = __builtin_amdgcn_wmma_f32_16x16x4_f32(false, a[k], false, b3, (short)0, acc3, false, false);
    }

    v8f accs[4] = {acc0, acc1, acc2, acc3};
#pragma unroll
    for (int nt = 0; nt < 4; ++nt) {
      float bias = Bv[e * DFEAT + nt * 16 + m];
#pragma unroll
      for (int r = 0; r < 8; ++r) {
        int row = r + 8 * half;
        if ((rows >> row) & 1u) {
          out[(size_t)tokrow[r] * DFEAT + nt * 16 + m] = accs[nt][r] + bias;
        }
      }
    }
  }
}

// ---------------- host-side launcher ----------------

extern "C" void kernel_launch(void* const* d_in, const int* in_sizes, int n_in,
                              void* d_out, int out_size, void* d_ws, size_t ws_size,
                              hipStream_t stream) {
  const float* X    = (const float*)d_in[0];   // [N, 64]
  const int*   inds = (const int*)  d_in[1];   // [N]
  const float* W    = (const float*)d_in[2];   // [E, 64, 64]
  const float* Bv   = (const float*)d_in[3];   // [E, 64]
  float*       out  = (float*)d_out;

  const int n = in_sizes[1];
  const int E = in_sizes[3] / DFEAT;

  int* ws      = (int*)d_ws;
  int* counts  = ws;        // 16 ints
  int* offsets = ws + 16;   // 16 ints
  int* cursors = ws + 32;   // 16 ints
  int* perm    = ws + 64;   // n ints

  hipLaunchKernelGGL(zero_meta_kernel, dim3(1), dim3(64), 0, stream, ws);
  hipLaunchKernelGGL(count_kernel, dim3((n + 255) / 256), dim3(256), 0, stream,
                     inds, counts, n);
  hipLaunchKernelGGL(scan_kernel, dim3(1), dim3(1), 0, stream,
                     counts, offsets, cursors, E);
  hipLaunchKernelGGL(scatter_kernel, dim3((n + 255) / 256), dim3(256), 0, stream,
                     inds, cursors, perm, n);
  hipLaunchKernelGGL(moe_wmma_kernel, dim3((n + 63) / 64), dim3(128), 0, stream,
                     X, inds, W, Bv, perm, out, n);
}